// Word2Vec_neg_sampling_21801253994630
// MI455X (gfx1250) — compile-verified
//
#include <hip/hip_runtime.h>
#include <math.h>

typedef __attribute__((ext_vector_type(2))) float v2f;
typedef __attribute__((ext_vector_type(8))) float v8f;

#define B_SZ   16384
#define NEG_K  10
#define E_DIM  128
#define TILES  (B_SZ / 16)   // 1024

// Extract element (lane&7) of an 8-wide f32 accumulator without dynamic
// vector indexing (lowers to a short v_cndmask chain).
__device__ __forceinline__ float diag_extract(v8f v, int lane) {
  switch (lane & 7) {
    case 0:  return v[0];
    case 1:  return v[1];
    case 2:  return v[2];
    case 3:  return v[3];
    case 4:  return v[4];
    case 5:  return v[5];
    case 6:  return v[6];
    default: return v[7];
  }
}

// Numerically stable log_sigmoid(x) = min(x,0) - log1p(exp(-|x|))
__device__ __forceinline__ float log_sigmoid(float x) {
  return fminf(x, 0.0f) - log1pf(expf(-fabsf(x)));
}

__global__ __launch_bounds__(32)
void w2v_sgns_tile_kernel(const int*   __restrict__ input_word,
                          const int*   __restrict__ context_word,
                          const int*   __restrict__ noise_words,
                          const float* __restrict__ W_in,
                          const float* __restrict__ W_ctx,
                          float*       __restrict__ partial)
{
  const int tile = blockIdx.x;        // 16 batch rows per tile
  const int lane = threadIdx.x;       // wave32
  const int m    = lane & 15;         // M (A rows) == N (B cols) slot
  const int half = lane >> 4;         // K sub-pair select per WMMA f32 layout
  const int row  = tile * 16 + m;     // batch row handled by this lane slot

  // ---- B operand: center^T, column n = W_in[input_word[row_n]] ----
  // Lane layout for 32-bit B 4x16: lanes 0-15 hold (K=0,1), lanes 16-31 (K=2,3)
  const long  crow  = (long)input_word[row];
  const float* cbase = W_in + crow * E_DIM + half * 2;
  v2f breg[32];
#pragma unroll
  for (int ck = 0; ck < 32; ++ck)
    breg[ck] = *(const v2f*)(cbase + 4 * ck);   // 8B aligned global_load_b64

  const bool diagLane = (lane < 8) || (lane >= 24);
  float lossAcc = 0.0f;

  // j==0: positive (ctx), j=1..10: negatives (noise). Same A layout as B.
  for (int j = 0; j < 11; ++j) {
    const long arow = (j == 0) ? (long)context_word[row]
                               : (long)noise_words[row * NEG_K + (j - 1)];
    const float* abase = W_ctx + arow * E_DIM + half * 2;

    v2f areg[32];
#pragma unroll
    for (int ck = 0; ck < 32; ++ck)
      areg[ck] = *(const v2f*)(abase + 4 * ck);

    v8f acc = {};
#pragma unroll
    for (int ck = 0; ck < 32; ++ck)
      acc = __builtin_amdgcn_wmma_f32_16x16x4_f32(
                /*neg_a=*/false, areg[ck],
                /*neg_b=*/false, breg[ck],
                /*c_mod=*/(short)0, acc,
                /*reuse_a=*/false, /*reuse_b=*/false);

    // Diagonal of the 16x16 product = per-row dot products.
    // m<8  -> (lane m,   elem m);  m>=8 -> (lane m+16, elem m-8)
    if (diagLane) {
      float d = diag_extract(acc, lane);
      float x = (j == 0) ? d : -d;           // negatives use -u_k . v_c
      lossAcc += log_sigmoid(x);
    }
  }

  // Wave-level butterfly reduce (non-diag lanes contribute 0).
  for (int off = 16; off >= 1; off >>= 1)
    lossAcc += __shfl_xor(lossAcc, off, 32);

  if (lane == 0) partial[tile] = lossAcc;
}

__global__ __launch_bounds__(256)
void w2v_reduce_kernel(const float* __restrict__ partial,
                       float*       __restrict__ out)
{
  __shared__ float s[256];
  const int t = threadIdx.x;
  float v = partial[t] + partial[t + 256] + partial[t + 512] + partial[t + 768];
  s[t] = v;
  __syncthreads();
  for (int off = 128; off > 0; off >>= 1) {
    if (t < off) s[t] += s[t + off];
    __syncthreads();
  }
  if (t == 0) out[0] = -s[0] / (float)B_SZ;   // -(mean of per-row losses)
}

extern "C" void kernel_launch(void* const* d_in, const int* in_sizes, int n_in,
                              void* d_out, int out_size, void* d_ws, size_t ws_size,
                              hipStream_t stream)
{
  const int*   input_word   = (const int*)d_in[0];
  const int*   context_word = (const int*)d_in[1];
  const int*   noise_words  = (const int*)d_in[2];
  const float* W_in         = (const float*)d_in[3];
  const float* W_ctx        = (const float*)d_in[4];
  float* out     = (float*)d_out;
  float* partial = (float*)d_ws;   // TILES floats = 4 KB scratch

  w2v_sgns_tile_kernel<<<TILES, 32, 0, stream>>>(
      input_word, context_word, noise_words, W_in, W_ctx, partial);
  w2v_reduce_kernel<<<1, 256, 0, stream>>>(partial, out);
}